// PY_GCG_NET_26508538151144
// MI455X (gfx1250) — compile-verified
//
#include <hip/hip_runtime.h>
#include <math.h>

#define NNODES 50000
#define NEDGES 800000
#define NGRAPHS 512

typedef __attribute__((ext_vector_type(2))) float v2f;
typedef __attribute__((ext_vector_type(8))) float v8f;

// ---------------- elementwise / utility kernels ----------------

__global__ void k_zero4(float4* __restrict__ p, int n4) {
    int i = blockIdx.x * blockDim.x + threadIdx.x;
    if (i < n4) p[i] = make_float4(0.f, 0.f, 0.f, 0.f);
}

__global__ void k_degree(const int* __restrict__ dst, float* __restrict__ deg, int E) {
    int i = blockIdx.x * blockDim.x + threadIdx.x;
    if (i < E) atomicAdd(&deg[dst[i]], 1.0f);
}

__global__ void k_dinv(const float* __restrict__ deg, float* __restrict__ dinv, int n) {
    int i = blockIdx.x * blockDim.x + threadIdx.x;
    if (i < n) dinv[i] = rsqrtf(deg[i] + 1.0f);   // deg = in-degree + 1 (self loop)
}

// ---------------- fp32 WMMA GEMM: C = A[MxK] @ B[KxNc] (+bias)(+relu) ------
// Compile-time K/Nc: K-loop fully unrolls into K/4 chained
// V_WMMA_F32_16X16X4_F32 ops; all addressing becomes shifts.
// ISA 7.12.2 32-bit layouts: A(16x4): lanes 0-15 hold K=0,1; lanes 16-31 K=2,3.
// B(4x16): N striped over lanes (lane&15), same K split. C/D: VGPR v holds
// row M = v + (lane<16 ? 0 : 8), col N = lane&15.

template <int K, int Nc, bool RELU>
__global__ void k_gemm_wmma(const float* __restrict__ A, const float* __restrict__ B,
                            const float* __restrict__ bias, float* __restrict__ C,
                            int M) {
    const int lane = threadIdx.x & 31;
    const int wave = threadIdx.x >> 5;
    const int wpb  = blockDim.x >> 5;
    constexpr int TILES_N = Nc >> 4;
    const int tile = blockIdx.x * wpb + wave;
    if (tile >= (M >> 4) * TILES_N) return;   // launch sized so this never splits a wave
    const int tm = tile / TILES_N;
    const int tn = tile % TILES_N;

    const int mrow = tm * 16 + (lane & 15);
    const int ncol = tn * 16 + (lane & 15);
    const int klo  = (lane < 16) ? 0 : 2;

    const float* __restrict__ Ap = A + (size_t)mrow * K + klo;
    const float* __restrict__ Bp = B + (size_t)klo * Nc + ncol;

    v8f acc = {};
#pragma unroll
    for (int k = 0; k < K; k += 4) {
        v2f a, b;
        a.x = Ap[k];
        a.y = Ap[k + 1];
        b.x = Bp[(size_t)k * Nc];
        b.y = Bp[(size_t)(k + 1) * Nc];
        acc = __builtin_amdgcn_wmma_f32_16x16x4_f32(
            /*neg_a=*/false, a, /*neg_b=*/false, b,
            /*c_mod=*/(short)0, acc, /*reuse_a=*/false, /*reuse_b=*/false);
    }

    const int row0 = tm * 16 + ((lane < 16) ? 0 : 8);
    const float bc = bias ? bias[ncol] : 0.0f;
#pragma unroll
    for (int v = 0; v < 8; ++v) {
        float val = acc[v] + bc;
        if (RELU) val = fmaxf(val, 0.0f);
        C[(size_t)(row0 + v) * Nc + ncol] = val;
    }
}

// ---------------- edge message scatter (L2-resident atomics) ---------------
// One wave per edge: src/dst/coef read once per wave; lane l handles
// channels [l*PER, l*PER+PER). out[dst] += h[src] * dinv[src]*dinv[dst].

template <int C>
__global__ void k_edge_scatter(const float* __restrict__ h, const int* __restrict__ src,
                               const int* __restrict__ dst, const float* __restrict__ dinv,
                               float* __restrict__ out, int E) {
    const int e = blockIdx.x * (blockDim.x >> 5) + (threadIdx.x >> 5);
    if (e >= E) return;
    const int lane = threadIdx.x & 31;
    const int s = src[e], d = dst[e];
    const float coef = dinv[s] * dinv[d];
    constexpr int PER = C / 32;               // 4 (C=128) or 2 (C=64)
    const float* hp = h   + (size_t)s * C + lane * PER;
    float*       op = out + (size_t)d * C + lane * PER;
    if (PER == 4) {
        const float4 v = *(const float4*)hp;
        atomicAdd(op + 0, v.x * coef);
        atomicAdd(op + 1, v.y * coef);
        atomicAdd(op + 2, v.z * coef);
        atomicAdd(op + 3, v.w * coef);
    } else {
        const float2 v = *(const float2*)hp;
        atomicAdd(op + 0, v.x * coef);
        atomicAdd(op + 1, v.y * coef);
    }
}

// out = relu(agg + hlin*dinv^2 + bias)   (self-loop term + bias + ReLU)
template <int C>
__global__ void k_finish_conv(const float* __restrict__ hlin, const float* __restrict__ agg,
                              const float* __restrict__ dinv, const float* __restrict__ bias,
                              float* __restrict__ out, int nN) {
    int i = blockIdx.x * blockDim.x + threadIdx.x;
    if (i >= nN * C) return;
    const int n = i / C, c = i & (C - 1);     // C is a power of two -> shift/mask
    const float di = dinv[n];
    const float v = agg[i] + hlin[i] * di * di + bias[c];
    out[i] = fmaxf(v, 0.0f);
}

// ---------------- BatchNorm (biased var, eps=1e-5) -------------------------

__global__ void k_bn_stats(const float* __restrict__ x, int rows, int cols,
                           float* __restrict__ mean, float* __restrict__ invstd) {
    __shared__ float ssum[256];
    __shared__ float ssq[256];
    const int c = blockIdx.x;
    float s = 0.0f, q = 0.0f;
    for (int r = threadIdx.x; r < rows; r += blockDim.x) {
        const float v = x[(size_t)r * cols + c];
        s += v; q += v * v;
    }
    ssum[threadIdx.x] = s; ssq[threadIdx.x] = q;
    __syncthreads();
    for (int o = blockDim.x >> 1; o > 0; o >>= 1) {
        if (threadIdx.x < o) {
            ssum[threadIdx.x] += ssum[threadIdx.x + o];
            ssq[threadIdx.x]  += ssq[threadIdx.x + o];
        }
        __syncthreads();
    }
    if (threadIdx.x == 0) {
        const float m = ssum[0] / (float)rows;
        const float var = ssq[0] / (float)rows - m * m;
        mean[c] = m;
        invstd[c] = rsqrtf(var + 1e-5f);
    }
}

template <int C>
__global__ void k_bn_norm(const float* __restrict__ x, const float* __restrict__ mean,
                          const float* __restrict__ invstd, const float* __restrict__ gamma,
                          const float* __restrict__ beta, float* __restrict__ out,
                          int rows) {
    int i = blockIdx.x * blockDim.x + threadIdx.x;
    if (i >= rows * C) return;
    const int c = i & (C - 1);
    out[i] = (x[i] - mean[c]) * invstd[c] * gamma[c] + beta[c];
}

// ---------------- global add pool ------------------------------------------

template <int C>
__global__ void k_pool(const float* __restrict__ h, const int* __restrict__ batch,
                       float* __restrict__ hg, int nN) {
    int i = blockIdx.x * blockDim.x + threadIdx.x;
    if (i >= nN * C) return;
    const int n = i / C, c = i & (C - 1);
    atomicAdd(&hg[(size_t)batch[n] * C + c], h[i]);
}

// ---------------- fc3 (64->3) + log_softmax --------------------------------

__global__ void k_head(const float* __restrict__ t2, const float* __restrict__ fW3,
                       const float* __restrict__ fb3, float* __restrict__ out, int G) {
    int g = blockIdx.x * blockDim.x + threadIdx.x;
    if (g >= G) return;
    float l[3];
#pragma unroll
    for (int j = 0; j < 3; ++j) {
        float s = fb3[j];
        for (int k = 0; k < 64; ++k) s += t2[(size_t)g * 64 + k] * fW3[k * 3 + j];
        l[j] = s;
    }
    const float mx  = fmaxf(l[0], fmaxf(l[1], l[2]));
    const float lse = mx + logf(expf(l[0] - mx) + expf(l[1] - mx) + expf(l[2] - mx));
    out[g * 3 + 0] = l[0] - lse;
    out[g * 3 + 1] = l[1] - lse;
    out[g * 3 + 2] = l[2] - lse;
}

// ---------------- launcher --------------------------------------------------

static inline int cdiv(long long a, int b) { return (int)((a + b - 1) / b); }

extern "C" void kernel_launch(void* const* d_in, const int* in_sizes, int n_in,
                              void* d_out, int out_size, void* d_ws, size_t ws_size,
                              hipStream_t stream) {
    const float* x    = (const float*)d_in[0];
    const int*   ei   = (const int*)  d_in[1];   // [2,E]: src = ei, dst = ei+E
    const int*   bat  = (const int*)  d_in[2];
    const float* W1   = (const float*)d_in[3];
    const float* b1   = (const float*)d_in[4];
    const float* g1   = (const float*)d_in[5];
    const float* be1  = (const float*)d_in[6];
    const float* W2   = (const float*)d_in[7];
    const float* b2   = (const float*)d_in[8];
    const float* g2   = (const float*)d_in[9];
    const float* be2  = (const float*)d_in[10];
    const float* fW1  = (const float*)d_in[11];
    const float* fb1  = (const float*)d_in[12];
    const float* g3   = (const float*)d_in[13];
    const float* be3  = (const float*)d_in[14];
    const float* fW2  = (const float*)d_in[15];
    const float* fb2  = (const float*)d_in[16];
    const float* fW3  = (const float*)d_in[17];
    const float* fb3  = (const float*)d_in[18];
    float* out = (float*)d_out;

    const int* src = ei;
    const int* dst = ei + NEDGES;

    // workspace layout (floats)
    float* ws = (float*)d_ws;
    size_t o = 0;
    float* deg   = ws + o; o += NNODES;
    float* dinv  = ws + o; o += NNODES;
    float* pad   = ws + o; o += 48;                      // align h1 to 16B
    float* h1    = ws + o; o += (size_t)NNODES * 128;    // x@W1, then relu(conv1)
    float* buf1  = ws + o; o += (size_t)NNODES * 128;    // agg1, then BN1 output
    float* h2    = ws + o; o += (size_t)NNODES * 64;     // h@W2, then relu(conv2)
    float* buf2  = ws + o; o += (size_t)NNODES * 64;     // agg2, then BN2 output
    float* hg    = ws + o; o += (size_t)NGRAPHS * 64;
    float* t1    = ws + o; o += (size_t)NGRAPHS * 64;
    float* t2    = ws + o; o += (size_t)NGRAPHS * 64;
    float* mean1 = ws + o; o += 128;
    float* istd1 = ws + o; o += 128;
    float* mean2 = ws + o; o += 64;
    float* istd2 = ws + o; o += 64;
    float* mean3 = ws + o; o += 64;
    float* istd3 = ws + o; o += 64;
    (void)pad;

    const int B = 256;

    // zero the scatter-add targets (must be re-zeroed every call)
    k_zero4<<<cdiv(NNODES / 4, B), B, 0, stream>>>((float4*)deg, NNODES / 4);
    k_zero4<<<cdiv((long long)NNODES * 32, B), B, 0, stream>>>((float4*)buf1, NNODES * 32);
    k_zero4<<<cdiv((long long)NNODES * 16, B), B, 0, stream>>>((float4*)buf2, NNODES * 16);
    k_zero4<<<cdiv(NGRAPHS * 16, B), B, 0, stream>>>((float4*)hg, NGRAPHS * 16);

    // degree + symmetric normalization
    k_degree<<<cdiv(NEDGES, B), B, 0, stream>>>(dst, deg, NEDGES);
    k_dinv<<<cdiv(NNODES, B), B, 0, stream>>>(deg, dinv, NNODES);

    // ---- GCN layer 1: 4 -> 128 ----
    // tiles = (50000/16)*(128/16) = 25000; 8 waves/block -> 3125 blocks (exact)
    k_gemm_wmma<4, 128, false><<<25000 / 8, 256, 0, stream>>>(x, W1, nullptr, h1, NNODES);
    k_edge_scatter<128><<<cdiv(NEDGES, 8), 256, 0, stream>>>(h1, src, dst, dinv, buf1, NEDGES);
    k_finish_conv<128><<<cdiv((long long)NNODES * 128, B), B, 0, stream>>>(h1, buf1, dinv, b1, h1, NNODES);
    k_bn_stats<<<128, 256, 0, stream>>>(h1, NNODES, 128, mean1, istd1);
    k_bn_norm<128><<<cdiv((long long)NNODES * 128, B), B, 0, stream>>>(h1, mean1, istd1, g1, be1, buf1, NNODES);

    // ---- GCN layer 2: 128 -> 64 ----
    // tiles = 3125*4 = 12500; 4 waves/block -> 3125 blocks (exact)
    k_gemm_wmma<128, 64, false><<<12500 / 4, 128, 0, stream>>>(buf1, W2, nullptr, h2, NNODES);
    k_edge_scatter<64><<<cdiv(NEDGES, 8), 256, 0, stream>>>(h2, src, dst, dinv, buf2, NEDGES);
    k_finish_conv<64><<<cdiv((long long)NNODES * 64, B), B, 0, stream>>>(h2, buf2, dinv, b2, h2, NNODES);
    k_bn_stats<<<64, 256, 0, stream>>>(h2, NNODES, 64, mean2, istd2);
    k_bn_norm<64><<<cdiv((long long)NNODES * 64, B), B, 0, stream>>>(h2, mean2, istd2, g2, be2, buf2, NNODES);

    // ---- global add pool ----
    k_pool<64><<<cdiv((long long)NNODES * 64, B), B, 0, stream>>>(buf2, bat, hg, NNODES);

    // ---- MLP head ----
    // fc1: 512x64 @ 64x64, +bias, relu. tiles = 32*4 = 128; 4 waves/block -> 32 blocks
    k_gemm_wmma<64, 64, true><<<128 / 4, 128, 0, stream>>>(hg, fW1, fb1, t1, NGRAPHS);
    k_bn_stats<<<64, 256, 0, stream>>>(t1, NGRAPHS, 64, mean3, istd3);
    k_bn_norm<64><<<cdiv(NGRAPHS * 64, B), B, 0, stream>>>(t1, mean3, istd3, g3, be3, t1, NGRAPHS);
    // fc2: +bias, relu (dropout is identity in eval)
    k_gemm_wmma<64, 64, true><<<128 / 4, 128, 0, stream>>>(t1, fW2, fb2, t2, NGRAPHS);
    // fc3 + log_softmax
    k_head<<<cdiv(NGRAPHS, 64), 64, 0, stream>>>(t2, fW3, fb3, out, NGRAPHS);
}